// HadamardTransform_79353815761168
// MI455X (gfx1250) — compile-verified
//
#include <hip/hip_runtime.h>

typedef __attribute__((ext_vector_type(2))) float v2f;
typedef __attribute__((ext_vector_type(4))) float v4f;
typedef __attribute__((ext_vector_type(8))) float v8f;
typedef __attribute__((ext_vector_type(4))) int   v4i;

#define FWHT_SIZE 4096          // 16 * 16 * 16
#define WAVES_PER_BLOCK 4
#define BLOCK_THREADS (WAVES_PER_BLOCK * 32)

#define AS_GLOBAL __attribute__((address_space(1)))
#define AS_LDS    __attribute__((address_space(3)))

#if __has_builtin(__builtin_amdgcn_global_store_async_from_lds_b128)
#define HAVE_ASYNC_STORE 1
#else
#define HAVE_ASYNC_STORE 0
#endif

// H16[r][c] = (-1)^popcount(r & c)  (Sylvester ordering)
__device__ __forceinline__ float hsign(int r, int c) {
    return (__builtin_popcount(r & c) & 1) ? -1.0f : 1.0f;
}

// Scheduling fence only: each row is wave-private and LDS ops are in-order
// per wave on CDNA5, so no hardware barrier is needed — this just prevents
// the compiler from reordering cross-lane LDS accesses across stages.
__device__ __forceinline__ void wave_fence() {
    __builtin_amdgcn_wave_barrier();
    asm volatile("" ::: "memory");
}

__device__ __forceinline__ void wait_dscnt0() {
#if __has_builtin(__builtin_amdgcn_s_wait_dscnt)
    __builtin_amdgcn_s_wait_dscnt(0);
#else
    asm volatile("s_wait_dscnt 0x0" ::: "memory");
#endif
}

__global__ __launch_bounds__(BLOCK_THREADS) void fwht4096_wmma(
        const float* __restrict__ x, const float* __restrict__ s,
        float* __restrict__ out, int nrows, float scale) {
    __shared__ float lds_all[WAVES_PER_BLOCK * FWHT_SIZE];   // 64 KB

    const int wave = threadIdx.x >> 5;
    const int lane = threadIdx.x & 31;
    const int row  = blockIdx.x * WAVES_PER_BLOCK + wave;
    if (row >= nrows) return;                 // wave-uniform exit

    float* __restrict__ L = lds_all + wave * FWHT_SIZE;
    const int n = lane & 15;                  // N column / within-tile M row
    const int h = lane >> 4;                  // lane half

    // B operand = H16; chunk c holds rows 4c..4c+3.
    // f32 4x16 B layout: VGPR v, lane-half h -> K = 4c + 2h + v, N = lane&15.
    v2f bf[4];   // stages 1,2: H16
    v2f bfs[4];  // stage 3:    H16 * (1/sqrt(4096))  -> output pre-scaled
#pragma unroll
    for (int c = 0; c < 4; ++c) {
        bf[c].x  = hsign(4 * c + 2 * h + 0, n);
        bf[c].y  = hsign(4 * c + 2 * h + 1, n);
        bfs[c].x = bf[c].x * scale;
        bfs[c].y = bf[c].y * scale;
    }

    // ---- Load row (b128 NT, coalesced), apply sign vector, stage to LDS ---
    // x is streamed once (256 MB) -> non-temporal so it doesn't thrash L2;
    // s (16 KB) is reused by every row -> regular temporal.
    const float* __restrict__ xrow = x + (size_t)row * FWHT_SIZE;
#pragma unroll 8
    for (int v = 0; v < FWHT_SIZE / 128; ++v) {
        const int idx = v * 128 + lane * 4;
        v4f xv = __builtin_nontemporal_load((const v4f*)(xrow + idx));
        v4f sv = *(const v4f*)(s + idx);
        *(v4f*)(L + idx) = xv * sv;
    }
    wave_fence();

    // ---- Stage 1: contract k.  A[m=(i,j)][kk=k], off = m*16 + kk ----------
    // Tile t reads and writes exactly L[t*256 .. t*256+255]  (in-place).
#pragma unroll
    for (int t = 0; t < 16; ++t) {
        const int mbase = (t * 16 + n) * 16;
        v8f d = {};
#pragma unroll
        for (int c = 0; c < 4; ++c) {
            const int kk0 = 4 * c + 2 * h;
            v2f a = *(const v2f*)(L + mbase + kk0);          // ds_load_b64
            d = __builtin_amdgcn_wmma_f32_16x16x4_f32(
                    false, a, false, bf[c], (short)0, d, false, false);
        }
#pragma unroll
        for (int r = 0; r < 8; ++r)                          // D[(i,j)][k']
            L[(t * 16 + r + 8 * h) * 16 + n] = d[r];
    }
    wave_fence();

    // ---- Stage 2: contract j.  A[m=(i,k')][kk=j], off = i*256 + kk*16 + k'
    // Tile t (i = t) reads and writes L[t*256 .. t*256+255]  (in-place).
#pragma unroll
    for (int t = 0; t < 16; ++t) {
        v8f d = {};
#pragma unroll
        for (int c = 0; c < 4; ++c) {
            const int kk0 = 4 * c + 2 * h;
            v2f a;
            a.x = L[t * 256 + (kk0 + 0) * 16 + n];
            a.y = L[t * 256 + (kk0 + 1) * 16 + n];
            d = __builtin_amdgcn_wmma_f32_16x16x4_f32(
                    false, a, false, bf[c], (short)0, d, false, false);
        }
#pragma unroll
        for (int r = 0; r < 8; ++r)                          // D[(i,k')][j']
            L[t * 256 + n * 16 + (r + 8 * h)] = d[r];
    }
    wave_fence();

    // ---- Stage 3: contract i.  A[m=(j',k')][kk=i], off = kk*256 + j'*16 + k'
    // B carries the 1/64 normalization, so LDS holds the final output after
    // this stage.  Tile t (j' = t) touches the stride-256 slab at column t.
#pragma unroll
    for (int t = 0; t < 16; ++t) {
        v8f d = {};
#pragma unroll
        for (int c = 0; c < 4; ++c) {
            const int kk0 = 4 * c + 2 * h;
            v2f a;
            a.x = L[(kk0 + 0) * 256 + t * 16 + n];
            a.y = L[(kk0 + 1) * 256 + t * 16 + n];
            d = __builtin_amdgcn_wmma_f32_16x16x4_f32(
                    false, a, false, bfs[c], (short)0, d, false, false);
        }
#pragma unroll
        for (int r = 0; r < 8; ++r)                          // D[(j',k')][i']
            L[n * 256 + t * 16 + (r + 8 * h)] = d[r];
    }
    wave_fence();

    // ---- Store row: LDS -> HBM -----------------------------------------
    float* __restrict__ orow = out + (size_t)row * FWHT_SIZE;
#if HAVE_ASYNC_STORE
    // Async DMA straight from LDS (no VGPR round-trip).  DScnt and ASYNCcnt
    // are independent, so explicitly drain stage-3 LDS writes first.
    wait_dscnt0();
#pragma unroll 8
    for (int v = 0; v < FWHT_SIZE / 128; ++v) {
        const int idx = v * 128 + lane * 4;
        __builtin_amdgcn_global_store_async_from_lds_b128(
                (AS_GLOBAL v4i*)(orow + idx), (AS_LDS v4i*)(L + idx), 0, 0);
    }
    // S_ENDPGM performs an implicit wait-idle, covering outstanding ASYNCcnt.
#else
    // Sync fallback (scale already folded into stage 3): b128 NT stores.
#pragma unroll 8
    for (int v = 0; v < FWHT_SIZE / 128; ++v) {
        const int idx = v * 128 + lane * 4;
        v4f yv = *(const v4f*)(L + idx);
        __builtin_nontemporal_store(yv, (v4f*)(orow + idx));
    }
#endif
}

extern "C" void kernel_launch(void* const* d_in, const int* in_sizes, int n_in,
                              void* d_out, int out_size, void* d_ws, size_t ws_size,
                              hipStream_t stream) {
    const float* x = (const float*)d_in[0];   // (ROWS, 4096) f32
    const float* s = (const float*)d_in[1];   // (4096,) f32 sign vector
    float* out = (float*)d_out;

    const int nrows   = in_sizes[0] / FWHT_SIZE;
    const int nblocks = (nrows + WAVES_PER_BLOCK - 1) / WAVES_PER_BLOCK;
    const float scale = 1.0f / 64.0f;         // 1/sqrt(4096)

    fwht4096_wmma<<<nblocks, BLOCK_THREADS, 0, stream>>>(x, s, out, nrows, scale);
}